// Model_76665166234133
// MI455X (gfx1250) — compile-verified
//
#include <hip/hip_runtime.h>

#define NNODES  500000
#define NEDGES  2000000
#define NCH     32
#define NLAYERS 8
#define NHIDDEN 1024
#define NGRAPHS 1024

typedef __attribute__((ext_vector_type(2))) float v2f;
typedef __attribute__((ext_vector_type(8))) float v8f;

// ---------------- utility kernels ----------------

__global__ void k_zero(float* __restrict__ p, int n) {
    int i = blockIdx.x * blockDim.x + threadIdx.x;
    if (i < n) p[i] = 0.0f;
}

__global__ void k_copy4(const float4* __restrict__ src, float4* __restrict__ dst, int n4) {
    int i = blockIdx.x * blockDim.x + threadIdx.x;
    if (i < n4) dst[i] = src[i];
}

// degree accumulation: norm[t] += 1, norm_t[s] += 1
__global__ void k_degree(const int* __restrict__ src, const int* __restrict__ tgt,
                         float* __restrict__ norm, float* __restrict__ norm_t, int ne) {
    int e = blockIdx.x * blockDim.x + threadIdx.x;
    if (e < ne) {
        atomicAdd(&norm[tgt[e]], 1.0f);
        atomicAdd(&norm_t[src[e]], 1.0f);
    }
}

__global__ void k_invert(float* __restrict__ p, int n) {
    int i = blockIdx.x * blockDim.x + threadIdx.x;
    if (i < n) p[i] = 1.0f / (1.0f + p[i]);
}

// x[i, :] = emb[nodes[i], :]  (float4 per lane; 8 lanes per node row)
__global__ void k_embed(const int* __restrict__ nodes, const float4* __restrict__ emb4,
                        float4* __restrict__ x4, int total /* NNODES*8 */) {
    int i = blockIdx.x * blockDim.x + threadIdx.x;
    if (i < total) x4[i] = emb4[nodes[i >> 3] * 8 + (i & 7)];
}

// agg[t[e], :] += x[s[e], :]
// one lane per (edge, 4-channel group): b128 coalesced gather, 4 scalar f32 atomics
__global__ void k_scatter(const int* __restrict__ src, const int* __restrict__ tgt,
                          const float* __restrict__ x, float* __restrict__ agg,
                          int total /* NEDGES*8 */) {
    int i = blockIdx.x * blockDim.x + threadIdx.x;
    if (i < total) {
        int e = i >> 3, q = (i & 7) << 2;
        const float4 v = *(const float4*)(x + (size_t)src[e] * NCH + q);
        float* dst = agg + (size_t)tgt[e] * NCH + q;
        atomicAdd(dst + 0, v.x);
        atomicAdd(dst + 1, v.y);
        atomicAdd(dst + 2, v.z);
        atomicAdd(dst + 3, v.w);
    }
}

// ---------------- WMMA conv: out = base + relu((norm .* agg) @ W) ----------------
// One wave32 per 16-node tile. A = [16 x 32] scaled agg rows, B = W [32 x 32] (LDS-staged),
// D split into two 16x16 tiles (out channels 0-15 / 16-31), K chained in 8x K=4 steps.
__global__ void k_conv_wmma(const float* __restrict__ agg, const float* __restrict__ norm,
                            const float* __restrict__ base, const float* __restrict__ W,
                            float* __restrict__ out, int ntiles) {
    __shared__ float Ws[NCH * NCH];
    // cooperative stage of the 4KB weight matrix: 256 threads x one float4
    ((float4*)Ws)[threadIdx.x] = ((const float4*)W)[threadIdx.x];
    __syncthreads();

    int lane = threadIdx.x & 31;
    int tile = blockIdx.x * (blockDim.x >> 5) + (threadIdx.x >> 5);
    if (tile >= ntiles) return;           // wave-uniform exit: EXEC all-ones at WMMA
    int col  = lane & 15;                 // A row index == D column index
    int kh   = lane >> 4;                 // K-half selector (lanes 16-31 hold K+2)
    int node = tile * 16 + col;
    float nrm = norm[node];
    const float* arow = agg + (size_t)node * NCH;

    v8f acc0 = {}; v8f acc1 = {};
#pragma unroll
    for (int kb = 0; kb < 8; ++kb) {
        int k0 = kb * 4 + kh * 2;
        v2f a;  a.x  = nrm * arow[k0];           a.y  = nrm * arow[k0 + 1];
        v2f b0; b0.x = Ws[k0 * NCH + col];       b0.y = Ws[(k0 + 1) * NCH + col];
        v2f b1; b1.x = Ws[k0 * NCH + 16 + col];  b1.y = Ws[(k0 + 1) * NCH + 16 + col];
        acc0 = __builtin_amdgcn_wmma_f32_16x16x4_f32(false, a, false, b0, (short)0, acc0, false, false);
        acc1 = __builtin_amdgcn_wmma_f32_16x16x4_f32(false, a, false, b1, (short)0, acc1, false, false);
    }
#pragma unroll
    for (int v = 0; v < 8; ++v) {
        int m  = v + 8 * kh;                    // D row: VGPR v, lane-half kh
        size_t on = (size_t)(tile * 16 + m) * NCH;
        float r0 = acc0[v]; r0 = r0 > 0.0f ? r0 : 0.0f;
        float r1 = acc1[v]; r1 = r1 > 0.0f ? r1 : 0.0f;
        out[on + col]      = base[on + col]      + r0;
        out[on + 16 + col] = base[on + 16 + col] + r1;
    }
}

// ---------------- pooling ----------------

// g[batch[i], :] += x[i, :]   (float4 loads, 4 scalar atomics)
__global__ void k_pool(const int* __restrict__ batch, const float* __restrict__ x,
                       float* __restrict__ g, int total /* NNODES*8 */) {
    int i = blockIdx.x * blockDim.x + threadIdx.x;
    if (i < total) {
        int n = i >> 3, q = (i & 7) << 2;
        const float4 v = *(const float4*)(x + (size_t)n * NCH + q);
        float* dst = g + (size_t)batch[n] * NCH + q;
        atomicAdd(dst + 0, v.x);
        atomicAdd(dst + 1, v.y);
        atomicAdd(dst + 2, v.z);
        atomicAdd(dst + 3, v.w);
    }
}

__global__ void k_count(const int* __restrict__ batch, float* __restrict__ counts, int n) {
    int i = blockIdx.x * blockDim.x + threadIdx.x;
    if (i < n) atomicAdd(&counts[batch[i]], 1.0f);
}

__global__ void k_div(float* __restrict__ g, const float* __restrict__ counts, int total) {
    int i = blockIdx.x * blockDim.x + threadIdx.x;
    if (i < total) g[i] /= counts[i >> 5];
}

// ---------------- WMMA MLP: h = relu(g @ hidden_w + b), [1024x32]@[32x1024] ----------------
__global__ void k_mlp1_wmma(const float* __restrict__ g, const float* __restrict__ hw,
                            const float* __restrict__ hb, float* __restrict__ h) {
    int lane = threadIdx.x & 31;
    int tid_tile = blockIdx.x * (blockDim.x >> 5) + (threadIdx.x >> 5); // 0..4095
    int rt = tid_tile >> 6;   // row tile 0..63
    int ct = tid_tile & 63;   // col tile 0..63
    int r  = lane & 15;
    int kh = lane >> 4;
    const float* grow = g + (size_t)(rt * 16 + r) * NCH;

    v8f acc = {};
#pragma unroll
    for (int kb = 0; kb < 8; ++kb) {
        int k0 = kb * 4 + kh * 2;
        v2f a; a.x = grow[k0];                        a.y = grow[k0 + 1];
        v2f b; b.x = hw[k0 * NHIDDEN + ct * 16 + r];  b.y = hw[(k0 + 1) * NHIDDEN + ct * 16 + r];
        acc = __builtin_amdgcn_wmma_f32_16x16x4_f32(false, a, false, b, (short)0, acc, false, false);
    }
#pragma unroll
    for (int v = 0; v < 8; ++v) {
        int m = v + 8 * kh;
        int colg = ct * 16 + r;
        float val = acc[v] + hb[colg];
        h[(size_t)(rt * 16 + m) * NHIDDEN + colg] = val > 0.0f ? val : 0.0f;
    }
}

// ---------------- readout: out[g] = sum_j h[g,j]*out_w[j] ----------------
__global__ void k_out(const float* __restrict__ h, const float* __restrict__ w,
                      float* __restrict__ out) {
    __shared__ float red[256];
    int gid = blockIdx.x;
    float s = 0.0f;
    for (int j = threadIdx.x; j < NHIDDEN; j += 256)
        s += h[(size_t)gid * NHIDDEN + j] * w[j];
    red[threadIdx.x] = s;
    __syncthreads();
    for (int off = 128; off > 0; off >>= 1) {
        if (threadIdx.x < off) red[threadIdx.x] += red[threadIdx.x + off];
        __syncthreads();
    }
    if (threadIdx.x == 0) out[gid] = red[0];
}

// ---------------- driver ----------------

extern "C" void kernel_launch(void* const* d_in, const int* in_sizes, int n_in,
                              void* d_out, int out_size, void* d_ws, size_t ws_size,
                              hipStream_t stream) {
    const int*   nodes    = (const int*)d_in[0];
    const int*   sources  = (const int*)d_in[1];
    const int*   targets  = (const int*)d_in[2];
    const int*   batch    = (const int*)d_in[3];
    const float* emb      = (const float*)d_in[4];
    const float* conv_w   = (const float*)d_in[5];
    const float* hidden_w = (const float*)d_in[6];
    const float* hidden_b = (const float*)d_in[7];
    const float* out_w    = (const float*)d_in[8];
    float* out = (float*)d_out;

    float* ws     = (float*)d_ws;
    float* norm   = ws;                                  // [N]
    float* norm_t = ws + NNODES;                         // [N]
    float* xA     = ws + (size_t)2 * NNODES;             // [N,32]
    float* xB     = xA + (size_t)NNODES * NCH;           // [N,32]
    float* agg    = xB + (size_t)NNODES * NCH;           // [N,32]
    // reuse agg region after the conv layers:
    float* gbuf   = agg;                                 // [1024,32]
    float* counts = agg + NGRAPHS * NCH;                 // [1024]
    float* hbuf   = counts + NGRAPHS;                    // [1024,1024]

    const int TPB = 256;
    const int NV4 = NNODES * 8;        // node float4 count (4M)
    const int EV4 = NEDGES * 8;        // edge float4-lane count (16M)

    // norms: zero, accumulate degrees, invert (norm & norm_t are contiguous)
    k_zero  <<<(2 * NNODES + TPB - 1) / TPB, TPB, 0, stream>>>(norm, 2 * NNODES);
    k_degree<<<(NEDGES + TPB - 1) / TPB, TPB, 0, stream>>>(sources, targets, norm, norm_t, NEDGES);
    k_invert<<<(2 * NNODES + TPB - 1) / TPB, TPB, 0, stream>>>(norm, 2 * NNODES);

    // x = emb[nodes]
    k_embed<<<(NV4 + TPB - 1) / TPB, TPB, 0, stream>>>(nodes, (const float4*)emb, (float4*)xA, NV4);

    float* cur = xA;
    float* nxt = xB;
    const int ntiles = NNODES / 16;                 // 31250, exact
    const int convBlocks = (ntiles + 7) / 8;        // 8 waves per 256-thread block

    for (int l = 0; l < NLAYERS; ++l) {
        // direction 0: aggregate sources -> targets, weight conv_w[l,0]
        k_copy4  <<<(NV4 + TPB - 1) / TPB, TPB, 0, stream>>>((const float4*)cur, (float4*)agg, NV4);
        k_scatter<<<(EV4 + TPB - 1) / TPB, TPB, 0, stream>>>(sources, targets, cur, agg, EV4);
        k_conv_wmma<<<convBlocks, TPB, 0, stream>>>(agg, norm, cur,
                                                    conv_w + (size_t)(l * 2 + 0) * NCH * NCH, nxt, ntiles);
        // direction 1: aggregate targets -> sources, weight conv_w[l,1], accumulate into nxt
        k_copy4  <<<(NV4 + TPB - 1) / TPB, TPB, 0, stream>>>((const float4*)cur, (float4*)agg, NV4);
        k_scatter<<<(EV4 + TPB - 1) / TPB, TPB, 0, stream>>>(targets, sources, cur, agg, EV4);
        k_conv_wmma<<<convBlocks, TPB, 0, stream>>>(agg, norm_t, nxt,
                                                    conv_w + (size_t)(l * 2 + 1) * NCH * NCH, nxt, ntiles);
        float* t = cur; cur = nxt; nxt = t;
    }

    // mean pool per graph
    k_zero <<<(NGRAPHS * NCH + NGRAPHS + TPB - 1) / TPB, TPB, 0, stream>>>(gbuf, NGRAPHS * NCH + NGRAPHS);
    k_pool <<<(NV4 + TPB - 1) / TPB, TPB, 0, stream>>>(batch, cur, gbuf, NV4);
    k_count<<<(NNODES + TPB - 1) / TPB, TPB, 0, stream>>>(batch, counts, NNODES);
    k_div  <<<(NGRAPHS * NCH + TPB - 1) / TPB, TPB, 0, stream>>>(gbuf, counts, NGRAPHS * NCH);

    // MLP head: 4096 16x16 tiles, 8 waves/block -> 512 blocks
    k_mlp1_wmma<<<512, TPB, 0, stream>>>(gbuf, hidden_w, hidden_b, hbuf);
    k_out<<<NGRAPHS, TPB, 0, stream>>>(hbuf, out_w, out);
}